// Custom_MSELoss_38225208934465
// MI455X (gfx1250) — compile-verified
//
#include <hip/hip_runtime.h>
#include <math.h>

// ---------------------------------------------------------------------------
// Problem constants (reference: B=32, J=17, H=W=256)
// ---------------------------------------------------------------------------
#define B_      32
#define J_      17
#define HW_     65536          // 256*256
#define NBJ     (B_ * J_)      // 544 (b,j) slices
#define THREADS 1024           // 32 wave32s per block
#define WAVES   (THREADS / 32)

typedef float f32x4 __attribute__((ext_vector_type(4)));
typedef float v2f   __attribute__((ext_vector_type(2)));
typedef float v8f   __attribute__((ext_vector_type(8)));

// Strict-greater update (first occurrence preserved when indices are visited
// in increasing order within a lane). Branchless: v_cmp + 2x v_cndmask.
__device__ __forceinline__ void amaxStrict(float& v, int& i, float ov, int oi) {
    const bool t = ov > v;
    v = t ? ov : v;
    i = t ? oi : i;
}

// Cross-lane combine with min-index tie-break (associative + commutative, so
// butterfly-safe). Branchless: predicates combined with bitwise ops.
__device__ __forceinline__ void amaxTie(float& v, int& i, float ov, int oi) {
    const bool t = (ov > v) | ((ov == v) & (oi < i));
    v = t ? ov : v;
    i = t ? oi : i;
}

// ---------------------------------------------------------------------------
// Kernel 1: per-(b,j) reduction. One block per slice of 65536 floats.
//   - streams output/target with NT b128 loads (285 MB stream, no L2 reuse)
//   - per-lane: pk-fma accumulate (p-t)^2 into 2 partials, argmax p, argmax t
//   - wave sq-sum reduction via V_WMMA_F32_16X16X4_F32 against all-ones B
//   - argmax wave reduction via branchless shfl_xor butterflies
// ---------------------------------------------------------------------------
__global__ __launch_bounds__(THREADS)
void bj_reduce(const float* __restrict__ tgt,     // d_in[0]: target (gt)
               const float* __restrict__ outp,    // d_in[1]: output (pred)
               float* __restrict__ wsSq,          // [NBJ] sum of squared diffs
               int*   __restrict__ wsI0,          // [NBJ] argmax(pred)
               int*   __restrict__ wsI1) {        // [NBJ] argmax(gt)
    const int    bj   = blockIdx.x;
    const size_t base = (size_t)bj * (size_t)HW_;
    const f32x4* p4   = (const f32x4*)(outp + base);
    const f32x4* t4   = (const f32x4*)(tgt  + base);
    const int    tid  = threadIdx.x;

    float sqx = 0.0f, sqy = 0.0f;
    float mpV = -__builtin_inff(); int mpI = 0x7FFFFFFF;
    float mtV = -__builtin_inff(); int mtI = 0x7FFFFFFF;

    #pragma unroll
    for (int i = 0; i < HW_ / 4 / THREADS; ++i) {   // 16 iterations
        const int v = tid + i * THREADS;            // strictly increasing
        f32x4 p = __builtin_nontemporal_load(p4 + v);   // global_load_b128 NT
        f32x4 t = __builtin_nontemporal_load(t4 + v);
        const int e = v * 4;

        float d0 = p.x - t.x, d1 = p.y - t.y, d2 = p.z - t.z, d3 = p.w - t.w;
        sqx = fmaf(d0, d0, sqx); sqy = fmaf(d1, d1, sqy);
        sqx = fmaf(d2, d2, sqx); sqy = fmaf(d3, d3, sqy);

        amaxStrict(mpV, mpI, p.x, e + 0); amaxStrict(mpV, mpI, p.y, e + 1);
        amaxStrict(mpV, mpI, p.z, e + 2); amaxStrict(mpV, mpI, p.w, e + 3);
        amaxStrict(mtV, mtI, t.x, e + 0); amaxStrict(mtV, mtI, t.y, e + 1);
        amaxStrict(mtV, mtI, t.z, e + 2); amaxStrict(mtV, mtI, t.w, e + 3);
    }

    // ---- wave reduction of sq partials on the matrix pipe -----------------
    // A = 16x4 f32 (the wave's 64 per-lane partials), B = 4x16 all-ones:
    // D[m,n] = sum_k A[m,k]. In-lane sum of the 8 accumulator VGPRs gives
    // lane L (<16) = sum_{m=0..7} D[m,L], lane L+16 = sum_{m=8..15} D[m,L];
    // one xor-16 shuffle completes the wave total in every lane.
    v2f a;  a.x = sqx;  a.y = sqy;
    v2f bo; bo.x = 1.0f; bo.y = 1.0f;
    v8f c = {};
    c = __builtin_amdgcn_wmma_f32_16x16x4_f32(false, a, false, bo,
                                              (short)0, c, false, false);
    float s = c[0] + c[1] + c[2] + c[3] + c[4] + c[5] + c[6] + c[7];
    float waveSq = s + __shfl_xor(s, 16, 32);

    // ---- wave reduction of the two argmaxes (branchless butterflies) ------
    #pragma unroll
    for (int off = 16; off > 0; off >>= 1) {
        float ov = __shfl_xor(mpV, off, 32); int oi = __shfl_xor(mpI, off, 32);
        amaxTie(mpV, mpI, ov, oi);
        ov = __shfl_xor(mtV, off, 32);       oi = __shfl_xor(mtI, off, 32);
        amaxTie(mtV, mtI, ov, oi);
    }

    // ---- cross-wave reduction through LDS (32 wave partials) --------------
    __shared__ float sSq[WAVES];
    __shared__ float sPV[WAVES]; __shared__ int sPI[WAVES];
    __shared__ float sTV[WAVES]; __shared__ int sTI[WAVES];
    const int wave = tid >> 5, lane = tid & 31;
    if (lane == 0) {
        sSq[wave] = waveSq;
        sPV[wave] = mpV; sPI[wave] = mpI;
        sTV[wave] = mtV; sTI[wave] = mtI;
    }
    __syncthreads();
    if (wave == 0) {
        float q  = sSq[lane];
        float pv = sPV[lane]; int pi = sPI[lane];
        float tv = sTV[lane]; int ti = sTI[lane];
        #pragma unroll
        for (int off = 16; off > 0; off >>= 1) {
            q += __shfl_xor(q, off, 32);
            float ov = __shfl_xor(pv, off, 32); int oi = __shfl_xor(pi, off, 32);
            amaxTie(pv, pi, ov, oi);
            ov = __shfl_xor(tv, off, 32);       oi = __shfl_xor(ti, off, 32);
            amaxTie(tv, ti, ov, oi);
        }
        if (lane == 0) { wsSq[bj] = q; wsI0[bj] = pi; wsI1[bj] = ti; }
    }
}

// ---------------------------------------------------------------------------
// Kernel 2: epilogue on one block of 544 threads (17 waves).
//   mse[j] = (1/(B*HW)) * sum_b sqsum[b,j]
//   dist[b,j] = sqrt((i0/256 - i1/256)^2 + (i0%256 - i1%256)^2)
//   loss = sum(dist * mse[j]) / J
// ---------------------------------------------------------------------------
__global__ __launch_bounds__(NBJ)
void finalize(const float* __restrict__ wsSq,
              const int*   __restrict__ wsI0,
              const int*   __restrict__ wsI1,
              float*       __restrict__ dOut) {
    __shared__ float mse[J_];
    const int tid = threadIdx.x;

    if (tid < J_) {
        float s = 0.0f;
        #pragma unroll
        for (int b = 0; b < B_; ++b) s += wsSq[b * J_ + tid];
        mse[tid] = s * (1.0f / ((float)B_ * (float)HW_));
    }
    __syncthreads();

    const int j  = tid % J_;                 // bj = b*J + j (J fastest)
    const int i0 = wsI0[tid];
    const int i1 = wsI1[tid];
    const int dx = (i0 >> 8) - (i1 >> 8);    // idx // 256
    const int dy = (i0 & 255) - (i1 & 255);  // idx %  256
    float val = sqrtf((float)(dx * dx + dy * dy)) * mse[j];

    #pragma unroll
    for (int off = 16; off > 0; off >>= 1) val += __shfl_xor(val, off, 32);

    __shared__ float part[NBJ / 32];         // 17 wave partials
    const int wave = tid >> 5, lane = tid & 31;
    if (lane == 0) part[wave] = val;
    __syncthreads();
    if (tid == 0) {
        float s = 0.0f;
        #pragma unroll
        for (int w = 0; w < NBJ / 32; ++w) s += part[w];
        dOut[0] = s / (float)J_;
    }
}

// ---------------------------------------------------------------------------
extern "C" void kernel_launch(void* const* d_in, const int* in_sizes, int n_in,
                              void* d_out, int out_size, void* d_ws, size_t ws_size,
                              hipStream_t stream) {
    const float* tgt  = (const float*)d_in[0];   // "target"
    const float* outp = (const float*)d_in[1];   // "output" (pred)

    float* wsSq = (float*)d_ws;                  // [544] f32
    int*   wsI0 = (int*)(wsSq + NBJ);            // [544] i32
    int*   wsI1 = wsI0 + NBJ;                    // [544] i32

    bj_reduce<<<NBJ, THREADS, 0, stream>>>(tgt, outp, wsSq, wsI0, wsI1);
    finalize<<<1, NBJ, 0, stream>>>(wsSq, wsI0, wsI1, (float*)d_out);
}